// PixelContrastLoss2_49503793054187
// MI455X (gfx1250) — compile-verified
//
#include <hip/hip_runtime.h>
#include <math.h>

// ---------------- problem constants (from reference setup_inputs) ------------
#define BB        2
#define FD        128
#define DD        16
#define HH        128
#define WW        128
#define SS        (DD * HH * WW)          // 262144 spatial per batch
#define NTOT      (BB * SS)               // 524288 pixels
#define NCLS      14
#define C13       (NCLS - 1)              // 13 sampled classes
#define NVIEW     50
#define NHARD     25
#define NEASY     25
#define VPAD      64                      // views padded to 4 tiles of 16
#define QLEN      4096
#define MNEG      2048                    // queue[:, 2048:, :]
#define NCHUNK    16                      // m-chunks per class in K4
#define TPC       8                       // m-tiles per chunk (16*8*16 = 2048)
#define TEMP_INV  10.0f
#define THRESH    0.3f

// ---------------- workspace layout (float units) -----------------------------
#define OFF_SUMS  0                               // 14*128
#define OFF_CNTS  (OFF_SUMS + NCLS * FD)          // 14
#define OFF_PN    1808                            // 13*128 normalized class means
#define OFF_AN    (OFF_PN + C13 * FD)             // 13*64*128 normalized views
#define OFF_INVN  (OFF_AN + C13 * VPAD * FD)      // 13*2048 inverse neg norms
#define OFF_NSUM  (OFF_INVN + C13 * MNEG)         // 13*64 exp-sum accumulators
#define OFF_SEL   (OFF_NSUM + C13 * VPAD)         // 13*50 ints

typedef __attribute__((ext_vector_type(2))) float v2f;
typedef __attribute__((ext_vector_type(8))) float v8f;

// ---------------- K0: zero accumulators --------------------------------------
__global__ void k_zero(float* ws) {
  int t = blockIdx.x * blockDim.x + threadIdx.x;
  if (t < OFF_PN) ws[t] = 0.0f;                       // sums + cnts (+pad)
  if (t < C13 * VPAD) ws[OFF_NSUM + t] = 0.0f;        // neg_sum
}

// ---------------- K1: ordered smallest-index selection -----------------------
// One wave per (class, hard/easy). Collect first k matching indices ascending.
__global__ void k_select(const int* __restrict__ lab, const int* __restrict__ pred,
                         const float* __restrict__ prob, int* __restrict__ sel) {
  int blk  = blockIdx.x;          // 0..25
  int c    = 1 + (blk >> 1);
  int type = blk & 1;             // 0 = hard, 1 = easy
  int lane = threadIdx.x;         // 32 threads (one wave)
  int k    = type ? NEASY : NHARD;
  int* dst = sel + (c - 1) * NVIEW + (type ? NHARD : 0);
  if (lane < k) dst[lane] = NTOT - 1;   // emulate JAX clamped OOB gather f[N]->f[N-1]
  int found = 0;
  for (int base = 0; base < NTOT && found < k; base += 32) {
    int n = base + lane;
    bool m = false;
    if (prob[n] > THRESH && pred[n] == c) {
      bool same = (lab[n] == c);
      m = type ? same : !same;
    }
    unsigned int mask = (unsigned int)__ballot(m);
    int rank = found + __popc(mask & ((1u << lane) - 1u));
    if (m && rank < k) dst[rank] = n;
    found += __popc(mask);
  }
}

// ---------------- K2: masked per-class feature sums + counts -----------------
// Block handles 1024 contiguous spatial positions (4 per thread, b128 loads),
// all 128 feature planes. This is the dominant 256 MB streaming pass.
__global__ void k_class_sums(const float* __restrict__ feats, const int* __restrict__ lab,
                             const float* __restrict__ prob,
                             float* __restrict__ sums, float* __restrict__ cnts) {
  __shared__ float lsum[NCLS * FD];   // 7 KB
  __shared__ float lcnt[NCLS];
  int t = threadIdx.x;                // 256
  for (int i = t; i < NCLS * FD; i += 256) lsum[i] = 0.0f;
  if (t < NCLS) lcnt[t] = 0.0f;
  int blocksPerBatch = SS / 1024;     // 256
  int b     = blockIdx.x / blocksPerBatch;
  int sbase = (blockIdx.x % blocksPerBatch) * 1024;
  int s0 = sbase + 4 * t;
  int n0 = b * SS + s0;
  float4 pv = *(const float4*)(prob + n0);
  int4   lv = *(const int4*)(lab + n0);
  int cls[4];
  cls[0] = (pv.x > THRESH) ? lv.x : -1;
  cls[1] = (pv.y > THRESH) ? lv.y : -1;
  cls[2] = (pv.z > THRESH) ? lv.z : -1;
  cls[3] = (pv.w > THRESH) ? lv.w : -1;
  __syncthreads();
  #pragma unroll
  for (int u = 0; u < 4; ++u)
    if (cls[u] >= 0) atomicAdd(&lcnt[cls[u]], 1.0f);
  const float* fb = feats + (size_t)b * FD * SS + s0;
  for (int d = 0; d < FD; ++d) {
    float4 v = *(const float4*)(fb + (size_t)d * SS);   // coalesced b128
    if (cls[0] >= 0) atomicAdd(&lsum[cls[0] * FD + d], v.x);
    if (cls[1] >= 0) atomicAdd(&lsum[cls[1] * FD + d], v.y);
    if (cls[2] >= 0) atomicAdd(&lsum[cls[2] * FD + d], v.z);
    if (cls[3] >= 0) atomicAdd(&lsum[cls[3] * FD + d], v.w);
  }
  __syncthreads();
  for (int i = t; i < NCLS * FD; i += 256) atomicAdd(&sums[i], lsum[i]);
  if (t < NCLS) atomicAdd(&cnts[t], lcnt[t]);
}

// ---------------- K3a: gather selected views, l2-normalize, pad to 64 --------
__global__ void k_build_a(const float* __restrict__ feats, const int* __restrict__ sel,
                          float* __restrict__ a_norm) {
  __shared__ float red[4];
  int blk = blockIdx.x;               // 13*64
  int c = blk / VPAD, v = blk % VPAD;
  int t = threadIdx.x;                // 128
  float* out = a_norm + ((size_t)(c * VPAD + v)) * FD;
  if (v >= NVIEW) { out[t] = 0.0f; return; }          // block-uniform
  int n = sel[c * NVIEW + v];
  int b = n / SS, s = n % SS;
  float val = feats[(size_t)b * FD * SS + (size_t)t * SS + s];
  float ss = val * val;
  for (int off = 16; off > 0; off >>= 1) ss += __shfl_xor(ss, off, 32);
  if ((t & 31) == 0) red[t >> 5] = ss;
  __syncthreads();
  float rn = rsqrtf(red[0] + red[1] + red[2] + red[3]);
  out[t] = val * rn;
}

// ---------------- K3b: class-mean positives, l2-normalize --------------------
__global__ void k_build_p(const float* __restrict__ sums, const float* __restrict__ cnts,
                          float* __restrict__ p_norm) {
  __shared__ float red[4];
  int c = blockIdx.x;                 // 0..12 -> class c+1
  int t = threadIdx.x;                // 128
  float val = sums[(c + 1) * FD + t] / cnts[c + 1];
  float ss = val * val;
  for (int off = 16; off > 0; off >>= 1) ss += __shfl_xor(ss, off, 32);
  if ((t & 31) == 0) red[t >> 5] = ss;
  __syncthreads();
  float rn = rsqrtf(red[0] + red[1] + red[2] + red[3]);
  p_norm[c * FD + t] = val * rn;
}

// ---------------- K3c: inverse norms of negative queue rows ------------------
__global__ void k_invn(const float* __restrict__ queue, float* __restrict__ invn) {
  __shared__ float red[4];
  int blk = blockIdx.x;               // 13*2048
  int c = blk / MNEG, m = blk % MNEG;
  int t = threadIdx.x;                // 128
  const float* row = queue + ((size_t)(c + 1) * QLEN + MNEG + m) * FD;
  float val = row[t];
  float ss = val * val;
  for (int off = 16; off > 0; off >>= 1) ss += __shfl_xor(ss, off, 32);
  if ((t & 31) == 0) red[t >> 5] = ss;
  __syncthreads();
  if (t == 0) invn[c * MNEG + m] = rsqrtf(red[0] + red[1] + red[2] + red[3]);
}

// ---- K4 helper: stage one 16x128 B tile into LDS, pre-scaled by 1/||row|| ---
__device__ __forceinline__ void stage_tile(float* dst, const float* __restrict__ queue,
                                           const float* __restrict__ invn,
                                           int c, int mt, int t) {
  #pragma unroll
  for (int i = t; i < 512; i += 128) {        // 512 float4 = 16x128 floats
    int row  = i >> 5;                        // 0..15
    int col4 = i & 31;                        // 0..31
    const float4* src =
        (const float4*)(queue + ((size_t)(c + 1) * QLEN + MNEG + mt * 16 + row) * FD) + col4;
    float4 v = *src;                          // coalesced b128
    float inv = invn[c * MNEG + mt * 16 + row];
    v.x *= inv; v.y *= inv; v.z *= inv; v.w *= inv;
    *((float4*)(dst + row * FD) + col4) = v;
  }
}

// ---------------- K4: WMMA a x neg^T, exp, row-sum ---------------------------
// Block = (class, chunk of 8 m-tiles). 128 threads = 4 waves = 4 v-tiles.
// A (all 64 rows) staged once; B tiles double-buffered in LDS, reused 4x.
__global__ void k_wmma_negsum(const float* __restrict__ queue,
                              const float* __restrict__ a_norm,
                              const float* __restrict__ invn,
                              float* __restrict__ neg_sum) {
  __shared__ float As[VPAD * FD];       // 32 KB
  __shared__ float Bs[2 * 16 * FD];     // 16 KB double buffer
  int c     = blockIdx.x / NCHUNK;
  int chunk = blockIdx.x % NCHUNK;
  int t     = threadIdx.x;              // 128
  int w     = t >> 5;                   // wave id = v-tile
  int lane  = t & 31;
  int half  = lane >> 4;                // K-pair selector (ISA 16x4 f32 layout)
  int idx16 = lane & 15;                // A row (M) / B col (N)
  int koff  = half * 2;

  // stage all four A v-tiles (contiguous 8192 floats) + first B tile
  {
    const float4* asrc = (const float4*)(a_norm + (size_t)c * VPAD * FD);
    #pragma unroll
    for (int i = t; i < 2048; i += 128) ((float4*)As)[i] = asrc[i];
  }
  stage_tile(Bs, queue, invn, c, chunk * TPC, t);
  __syncthreads();

  // preload this wave's A fragment: 32 k-steps x v2f
  v2f areg[32];
  {
    const float* asrc = As + (w * 16 + idx16) * FD + koff;
    #pragma unroll
    for (int j = 0; j < 32; ++j) areg[j] = *(const v2f*)(asrc + 4 * j);
  }

  float nsum[8];
  #pragma unroll
  for (int r = 0; r < 8; ++r) nsum[r] = 0.0f;

  for (int j = 0; j < TPC; ++j) {                       // block-uniform loop
    int buf = j & 1;
    if (j + 1 < TPC)                                    // overlap next tile's loads
      stage_tile(Bs + (1 - buf) * 16 * FD, queue, invn, c, chunk * TPC + j + 1, t);

    const float* bsrc = Bs + buf * 16 * FD + idx16 * FD + koff;
    v8f acc = {};
    #pragma unroll
    for (int s = 0; s < 32; ++s) {                      // 32 independent ds_load_b64
      v2f b = *(const v2f*)(bsrc + 4 * s);
      acc = __builtin_amdgcn_wmma_f32_16x16x4_f32(
          /*neg_a=*/false, areg[s], /*neg_b=*/false, b,
          /*c_mod=*/(short)0, acc, /*reuse_a=*/false, /*reuse_b=*/false);
    }
    // acc[r] = D[M][N], M = r + 8*half, N = idx16; exp + reduce over N
    #pragma unroll
    for (int r = 0; r < 8; ++r) {
      float e = __expf(acc[r] * TEMP_INV);
      e += __shfl_xor(e, 1, 32);
      e += __shfl_xor(e, 2, 32);
      e += __shfl_xor(e, 4, 32);
      e += __shfl_xor(e, 8, 32);
      nsum[r] += e;
    }
    __syncthreads();    // next-tile staging done; all waves done reading buf
  }
  if (idx16 == 0) {
    #pragma unroll
    for (int r = 0; r < 8; ++r)
      atomicAdd(&neg_sum[c * VPAD + w * 16 + r + 8 * half], nsum[r]);
  }
}

// ---------------- K5: logits + log-sum-exp + mean ----------------------------
__global__ void k_final(const float* __restrict__ a_norm, const float* __restrict__ p_norm,
                        const float* __restrict__ neg_sum, float* __restrict__ out) {
  __shared__ float red[8];
  int t = threadIdx.x;                // 256
  float local = 0.0f;
  for (int i = t; i < C13 * NVIEW; i += 256) {
    int c = i / NVIEW, v = i % NVIEW;
    const float* ar = a_norm + ((size_t)(c * VPAD + v)) * FD;
    const float* pr = p_norm + c * FD;
    float dot = 0.0f;
    for (int d = 0; d < FD; ++d) dot += ar[d] * pr[d];
    float logit = dot * TEMP_INV;
    local += -logit + logf(__expf(logit) + neg_sum[c * VPAD + v]);
  }
  for (int off = 16; off > 0; off >>= 1) local += __shfl_xor(local, off, 32);
  if ((t & 31) == 0) red[t >> 5] = local;
  __syncthreads();
  if (t == 0) {
    float tot = 0.0f;
    for (int w = 0; w < 8; ++w) tot += red[w];
    out[0] = tot / (float)(C13 * NVIEW);
  }
}

// ---------------- host-side launch -------------------------------------------
extern "C" void kernel_launch(void* const* d_in, const int* in_sizes, int n_in,
                              void* d_out, int out_size, void* d_ws, size_t ws_size,
                              hipStream_t stream) {
  (void)in_sizes; (void)n_in; (void)out_size; (void)ws_size;
  const float* feats  = (const float*)d_in[0];
  const int*   labels = (const int*)d_in[1];
  const int*   predic = (const int*)d_in[2];
  const float* prob   = (const float*)d_in[3];
  const float* queue  = (const float*)d_in[4];

  float* ws      = (float*)d_ws;
  float* sums    = ws + OFF_SUMS;
  float* cnts    = ws + OFF_CNTS;
  float* p_norm  = ws + OFF_PN;
  float* a_norm  = ws + OFF_AN;
  float* invn    = ws + OFF_INVN;
  float* neg_sum = ws + OFF_NSUM;
  int*   sel     = (int*)(ws + OFF_SEL);
  float* out     = (float*)d_out;

  k_zero<<<8, 256, 0, stream>>>(ws);
  k_select<<<2 * C13, 32, 0, stream>>>(labels, predic, prob, sel);
  k_class_sums<<<NTOT / 1024, 256, 0, stream>>>(feats, labels, prob, sums, cnts);
  k_build_a<<<C13 * VPAD, 128, 0, stream>>>(feats, sel, a_norm);
  k_build_p<<<C13, 128, 0, stream>>>(sums, cnts, p_norm);
  k_invn<<<C13 * MNEG, 128, 0, stream>>>(queue, invn);
  k_wmma_negsum<<<C13 * NCHUNK, 128, 0, stream>>>(queue, a_norm, invn, neg_sum);
  k_final<<<1, 256, 0, stream>>>(a_norm, p_norm, neg_sum, out);
}